// LSTMNet_2207613190832
// MI455X (gfx1250) — compile-verified
//
#include <hip/hip_runtime.h>
#include <stdint.h>
#include <stddef.h>

// ---------------------------------------------------------------------------
// LSTM scan, T=4096, B=8192, D=2.  (B,T,2) output.
// Lane-per-batch recurrence (wave32), triple-buffered async global->LDS
// streaming of x (ASYNCcnt), prescaled SGPR weights, packed-f32 x-side z
// (off-chain), scalar h-side z + gate tail (no broadcast movs on the serial
// chain), gfx1250 v_tanh_f32 gates, transposed nontemporal b128 stores.
// Roofline: 536 MB @ 23.3 TB/s => ~23 us floor; binding constraint is the
// 4096-step serial dependency chain -> minimize per-step chain latency.
// ---------------------------------------------------------------------------

#define T_LEN 4096
#define B_LEN 8192
#define D_LEN 2
#define TT    16                 // timesteps per chunk
#define NT    32                 // threads per block = 1 wave
#define NBUF  3                  // async pipeline depth
#define NC    (T_LEN / TT)       // 256 chunks

typedef float v4f __attribute__((ext_vector_type(4)));
typedef float v2f __attribute__((ext_vector_type(2)));

// Hardware tanh (gfx1250 trans op).
#if __has_builtin(__builtin_amdgcn_tanhf)
__device__ __forceinline__ float htanh(float x) { return __builtin_amdgcn_tanhf(x); }
#elif __has_builtin(__builtin_amdgcn_tanh_f32)
__device__ __forceinline__ float htanh(float x) { return __builtin_amdgcn_tanh_f32(x); }
#else
__device__ __forceinline__ float htanh(float x) {
  float r;
  asm volatile("v_tanh_f32 %0, %1" : "=v"(r) : "v"(x));
  return r;
}
#endif

__global__ __launch_bounds__(NT, 1) void lstm_scan_kernel(
    const float* __restrict__ xin,   // [T, B, 2]
    const float* __restrict__ U,     // [2, 8]
    const float* __restrict__ V,     // [2, 8]
    const float* __restrict__ bias,  // [8]
    float* __restrict__ out)         // [B, T, 2]
{
  // Triple-buffered x staging: [buf][tt][lane][d] = 12 KB per block.
  __shared__ float shx[NBUF][TT][NT][D_LEN];

  const int tid = threadIdx.x;
  const int b   = blockIdx.x * NT + tid;   // batch element owned by this lane

  // Prescaled, wave-uniform weights (SGPRs).  Columns j: 0,1=f  2,3=i
  // 4,5=o  6,7=g.  f/i/o: sigmoid(z)=0.5*tanh(0.5z)+0.5 -> prescale 0.5;
  // g: tanh(z) -> prescale 1.0.
  // Packed x-side weights (off-chain pk_fma) ...
  v2f ua2[4], ub2[4], bz2[4];
  // ... scalar h-side weights (chain FMAs, no broadcast needed).
  float vaS[8], vbS[8];
#pragma unroll
  for (int G = 0; G < 4; ++G) {
    const float s = (G == 3) ? 1.0f : 0.5f;
    ua2[G] = (v2f){ s * U[2 * G],     s * U[2 * G + 1]     };
    ub2[G] = (v2f){ s * U[8 + 2 * G], s * U[8 + 2 * G + 1] };
    bz2[G] = (v2f){ s * bias[2 * G],  s * bias[2 * G + 1]  };
    vaS[2 * G]     = s * V[2 * G];
    vaS[2 * G + 1] = s * V[2 * G + 1];
    vbS[2 * G]     = s * V[8 + 2 * G];
    vbS[2 * G + 1] = s * V[8 + 2 * G + 1];
  }

  // Issue one chunk: TT async b64 loads, 8B/lane, lane-coalesced 256B/wave.
#define ISSUE_CHUNK(kk, bf)                                                    \
  {                                                                            \
    _Pragma("unroll")                                                          \
    for (int tt_ = 0; tt_ < TT; ++tt_) {                                       \
      const float* ga_ =                                                       \
          xin + ((size_t)((kk) * TT + tt_) * B_LEN + (size_t)b) * D_LEN;       \
      unsigned int la_ = (unsigned int)(uintptr_t)&shx[bf][tt_][tid][0];       \
      asm volatile("global_load_async_to_lds_b64 %0, %1, off"                  \
                   :: "v"(la_), "v"((unsigned long long)(uintptr_t)ga_)        \
                   : "memory");                                                \
    }                                                                          \
  }

  // Prologue: start streaming chunks 0 and 1.
  ISSUE_CHUNK(0, 0)
  ISSUE_CHUNK(1, 1)

  float h0 = 0.0f, h1 = 0.0f, c0 = 0.0f, c1 = 0.0f;

  for (int k = 0; k < NC; ++k) {
    const int buf = k % NBUF;

    if (k + 2 < NC) {
      // Keep 2 chunks in flight; chunk k has landed once <= 32 ops remain.
      ISSUE_CHUNK(k + 2, (k + 2) % NBUF)
      asm volatile("s_wait_asynccnt 32" ::: "memory");
    } else if (k + 1 < NC) {
      asm volatile("s_wait_asynccnt 16" ::: "memory");
    } else {
      asm volatile("s_wait_asynccnt 0" ::: "memory");
    }

    float hs[TT][2];

#pragma unroll
    for (int tt = 0; tt < TT; ++tt) {
      const v2f x01 = *(const v2f*)&shx[buf][tt][tid][0];   // ds_load_b64
      const v2f x0b = (v2f){x01.x, x01.x};
      const v2f x1b = (v2f){x01.y, x01.y};

      // x-side of z: recurrence-independent, packed -> schedules off-chain.
      v2f zx[4];
#pragma unroll
      for (int G = 0; G < 4; ++G) {
        zx[G] = __builtin_elementwise_fma(
                    x1b, ub2[G], __builtin_elementwise_fma(x0b, ua2[G], bz2[G]));
      }

      // h-side + gates, fully scalar: h feeds FMAs directly (no movs on
      // chain), tanh feeds the affine directly.
      float t[8];
#pragma unroll
      for (int j = 0; j < 8; ++j) {
        const float zxj = (j & 1) ? zx[j >> 1].y : zx[j >> 1].x;
        const float zj  = __builtin_fmaf(h1, vbS[j],
                            __builtin_fmaf(h0, vaS[j], zxj));
        t[j] = htanh(zj);                       // v_tanh_f32
      }

      const float f0 = __builtin_fmaf(t[0], 0.5f, 0.5f);
      const float f1 = __builtin_fmaf(t[1], 0.5f, 0.5f);
      const float i0 = __builtin_fmaf(t[2], 0.5f, 0.5f);
      const float i1 = __builtin_fmaf(t[3], 0.5f, 0.5f);
      const float o0 = __builtin_fmaf(t[4], 0.5f, 0.5f);
      const float o1 = __builtin_fmaf(t[5], 0.5f, 0.5f);

      c0 = __builtin_fmaf(f0, c0, i0 * t[6]);
      c1 = __builtin_fmaf(f1, c1, i1 * t[7]);
      h0 = o0 * htanh(c0);
      h1 = o1 * htanh(c1);

      hs[tt][0] = h0;
      hs[tt][1] = h1;
    }

    // Transposed store: 16 steps * 2 floats = 128 contiguous bytes per lane,
    // as 8 nontemporal b128 stores (streaming output, never reread).
    float* op = out + ((size_t)b * T_LEN + (size_t)k * TT) * D_LEN;
#pragma unroll
    for (int q = 0; q < (TT * D_LEN) / 4; ++q) {
      v4f val = { hs[2 * q][0], hs[2 * q][1], hs[2 * q + 1][0], hs[2 * q + 1][1] };
      __builtin_nontemporal_store(val, ((v4f*)op) + q);
    }
  }
#undef ISSUE_CHUNK
}

extern "C" void kernel_launch(void* const* d_in, const int* in_sizes, int n_in,
                              void* d_out, int out_size, void* d_ws, size_t ws_size,
                              hipStream_t stream) {
  (void)in_sizes; (void)n_in; (void)out_size; (void)d_ws; (void)ws_size;
  const float* xin  = (const float*)d_in[0];  // [T, B, 2]
  const float* U    = (const float*)d_in[1];  // [2, 8]
  const float* V    = (const float*)d_in[2];  // [2, 8]
  const float* bias = (const float*)d_in[3];  // [8]
  float*       out  = (float*)d_out;          // [B, T, 2]

  dim3 grid(B_LEN / NT);   // 256 single-wave workgroups -> max WGP spread
  dim3 block(NT);
  lstm_scan_kernel<<<grid, block, 0, stream>>>(xin, U, V, bias, out);
}